// AttentionMechanism_45406394254132
// MI455X (gfx1250) — compile-verified
//
#include <hip/hip_runtime.h>
#include <math.h>

// Problem constants (match reference)
#define BB   16
#define TT   1024   // T_OUT
#define SS   1024   // T_IN
#define HH   256
#define HH2  512

typedef __attribute__((ext_vector_type(2))) float v2f;
typedef __attribute__((ext_vector_type(4))) float v4f;
typedef __attribute__((ext_vector_type(8))) float v8f;
typedef __attribute__((ext_vector_type(4))) int   v4i;

#if defined(__AMDGCN__) && __has_builtin(__builtin_amdgcn_global_load_async_to_lds_b128)
#define USE_ASYNC_BUILTIN 1
#endif

// Builtin takes generic typed pointers (per hipcc diagnostic: param is v4i*).
__device__ __forceinline__ v4i* as_v4i(const void* p) {
    return (v4i*)(uintptr_t)p;
}
__device__ __forceinline__ void wait_async0() {
#if defined(__AMDGCN__) && __has_builtin(__builtin_amdgcn_s_wait_asynccnt)
    __builtin_amdgcn_s_wait_asynccnt(0);
#else
    asm volatile("s_wait_asynccnt 0" ::: "memory");
#endif
}

// D = A(16x4 f32) * B(4x16 f32) + C, full f32 precision on the matrix core.
__device__ __forceinline__ v8f wmma_f32_k4(v2f a, v2f b, v8f c) {
    return __builtin_amdgcn_wmma_f32_16x16x4_f32(
        /*neg_a=*/false, a, /*neg_b=*/false, b,
        /*c_mod=*/(short)0, c, /*reuse_a=*/false, /*reuse_b=*/false);
}

__global__ __launch_bounds__(256)
void AttentionMechanism_45406394254132_kernel(
    const float* __restrict__ outp,         // [B, T, H]   "output" (queries)
    const float* __restrict__ ctx,          // [B, S, H]   "context" (keys/values)
    const unsigned char* __restrict__ mask, // [B, T, S] bool
    const float* __restrict__ W,            // [H, 2H]
    const float* __restrict__ bias,         // [H]
    float* __restrict__ out,                // [B, T, H]
    float* __restrict__ attn_out)           // [B, T, S]
{
    __shared__ float Qs[16 * HH];   // 16 KB : Q tile [16][H]
    __shared__ float Ss[16 * SS];   // 64 KB : scores -> probabilities [16][S]
    __shared__ float Ms[16 * HH];   // 16 KB : mix tile [16][H]

    const int b    = blockIdx.y;
    const int row0 = blockIdx.x * 16;
    const int tid  = threadIdx.x;
    const int wave = tid >> 5;          // 0..7
    const int lane = tid & 31;
    const int half = lane >> 4;         // 0 or 1 (lane group)
    const int l16  = lane & 15;

    // ---------------- Pass 0: stage Q tile into LDS (async copy path) ----------------
    {
        const float* Qg = outp + ((size_t)b * TT + row0) * HH;  // 16 contiguous rows
#ifdef USE_ASYNC_BUILTIN
        for (int i = tid * 4; i < 16 * HH; i += 256 * 4) {
            __builtin_amdgcn_global_load_async_to_lds_b128(
                as_v4i(Qg + i), as_v4i(&Qs[i]), /*offset=*/0, /*cpol=*/0);
        }
        wait_async0();
#else
        for (int i = tid; i < 16 * HH; i += 256) Qs[i] = Qg[i];
#endif
    }
    __syncthreads();

    const float* Kg = ctx + (size_t)b * SS * HH;

    // ---------------- Pass 1: S = Q @ K^T  (16 x 1024) ----------------
    // 64 S-tiles; each wave owns pairs (base, base+8) -> two independent
    // accumulator chains per wave (matrix-pipe ILP=2), shared A operand.
    for (int base = wave; base < SS / 16; base += 16) {
        const int colA = base * 16;
        const int colB = (base + 8) * 16;
        v8f acc0 = {}, acc1 = {};
        const float* qrow  = Qs + l16 * HH + half * 2;
        const float* krow0 = Kg + ((size_t)(colA + l16)) * HH + half * 2;
        const float* krow1 = Kg + ((size_t)(colB + l16)) * HH + half * 2;
        for (int k0 = 0; k0 < HH; k0 += 4) {
            v2f a  = *(const v2f*)(qrow  + k0);
            v2f b0 = *(const v2f*)(krow0 + k0);
            v2f b1 = *(const v2f*)(krow1 + k0);
            acc0 = wmma_f32_k4(a, b0, acc0);
            acc1 = wmma_f32_k4(a, b1, acc1);
        }
#pragma unroll
        for (int r = 0; r < 8; ++r) {
            Ss[(r + 8 * half) * SS + colA + l16] = acc0[r];
            Ss[(r + 8 * half) * SS + colB + l16] = acc1[r];
        }
    }
    __syncthreads();

    // ---------------- Pass 2: row softmax (+ mask), emit attn ----------------
    // 16 rows; each wave handles 2 rows with wave32 shfl reductions, b128 I/O.
    {
        const unsigned char* mbase = mask + ((size_t)b * TT + row0) * SS;
        for (int r = wave; r < 16; r += 8) {
            float* srow = Ss + r * SS;
            const unsigned* mrow = (const unsigned*)(mbase + (size_t)r * SS);
            float mx = -INFINITY;
            for (int c = lane * 4; c < SS; c += 128) {
                v4f s = *(const v4f*)(srow + c);
                unsigned m = __builtin_nontemporal_load(mrow + (c >> 2));
                s.x = (m & 0x000000ffu) ? -INFINITY : s.x;
                s.y = (m & 0x0000ff00u) ? -INFINITY : s.y;
                s.z = (m & 0x00ff0000u) ? -INFINITY : s.z;
                s.w = (m & 0xff000000u) ? -INFINITY : s.w;
                *(v4f*)(srow + c) = s;
                mx = fmaxf(mx, fmaxf(fmaxf(s.x, s.y), fmaxf(s.z, s.w)));
            }
#pragma unroll
            for (int off = 16; off > 0; off >>= 1)
                mx = fmaxf(mx, __shfl_xor(mx, off, 32));
            float sum = 0.0f;
            for (int c = lane * 4; c < SS; c += 128) {
                v4f s = *(const v4f*)(srow + c);
                v4f e;
                e.x = __expf(s.x - mx); e.y = __expf(s.y - mx);
                e.z = __expf(s.z - mx); e.w = __expf(s.w - mx);
                *(v4f*)(srow + c) = e;
                sum += (e.x + e.y) + (e.z + e.w);
            }
#pragma unroll
            for (int off = 16; off > 0; off >>= 1)
                sum += __shfl_xor(sum, off, 32);
            const float inv = 1.0f / sum;
            float* arow = attn_out + ((size_t)b * TT + row0 + r) * SS;
            for (int c = lane * 4; c < SS; c += 128) {
                v4f p = *(const v4f*)(srow + c);
                p *= inv;
                *(v4f*)(srow + c) = p;                            // P kept for pass 3
                __builtin_nontemporal_store(p, (v4f*)(arow + c)); // streamed past L2
            }
        }
    }
    __syncthreads();

    // ---------------- Pass 3: mix = P @ context  (16 x 256) ----------------
    // 16 tiles; each wave owns the pair (wave, wave+8).
    {
        const int colA = wave * 16;
        const int colB = colA + 128;
        v8f acc0 = {}, acc1 = {};
        const float* prow = Ss + l16 * SS + half * 2;
        const float* bb0  = Kg + colA + l16;   // B element (k,n) = ctx[k*H + n]
        const float* bb1  = Kg + colB + l16;
        for (int k0 = 0; k0 < SS; k0 += 4) {
            v2f a = *(const v2f*)(prow + k0);
            const size_t r0 = (size_t)(k0 + 2 * half) * HH;
            const size_t r1 = r0 + HH;
            v2f b0, b1;
            b0[0] = bb0[r0]; b0[1] = bb0[r1];
            b1[0] = bb1[r0]; b1[1] = bb1[r1];
            acc0 = wmma_f32_k4(a, b0, acc0);
            acc1 = wmma_f32_k4(a, b1, acc1);
        }
#pragma unroll
        for (int r = 0; r < 8; ++r) {
            Ms[(r + 8 * half) * HH + colA + l16] = acc0[r];
            Ms[(r + 8 * half) * HH + colB + l16] = acc1[r];
        }
    }
    __syncthreads();

    // ---------------- Pass 4: out = tanh([mix | Q] @ W^T + b) ----------------
    {
        const int colA = wave * 16;
        const int colB = colA + 128;
        v8f acc0 = {}, acc1 = {};
        const float* wrow0 = W + ((size_t)(colA + l16)) * HH2 + half * 2;  // B: W[n, k]
        const float* wrow1 = W + ((size_t)(colB + l16)) * HH2 + half * 2;
        // k in [0, 256): combined = mix
        const float* arow = Ms + l16 * HH + half * 2;
        for (int k0 = 0; k0 < HH; k0 += 4) {
            v2f a  = *(const v2f*)(arow  + k0);
            v2f b0 = *(const v2f*)(wrow0 + k0);
            v2f b1 = *(const v2f*)(wrow1 + k0);
            acc0 = wmma_f32_k4(a, b0, acc0);
            acc1 = wmma_f32_k4(a, b1, acc1);
        }
        // k in [256, 512): combined = Q
        const float* arow2 = Qs + l16 * HH + half * 2;
        for (int k0 = 0; k0 < HH; k0 += 4) {
            v2f a  = *(const v2f*)(arow2 + k0);
            v2f b0 = *(const v2f*)(wrow0 + HH + k0);
            v2f b1 = *(const v2f*)(wrow1 + HH + k0);
            acc0 = wmma_f32_k4(a, b0, acc0);
            acc1 = wmma_f32_k4(a, b1, acc1);
        }
        const float bnA = bias[colA + l16];
        const float bnB = bias[colB + l16];
#pragma unroll
        for (int r = 0; r < 8; ++r) {
            const int m = r + 8 * half;
            float* orow = out + ((size_t)b * TT + row0 + m) * HH;
            __builtin_nontemporal_store(tanhf(acc0[r] + bnA), orow + colA + l16);
            __builtin_nontemporal_store(tanhf(acc1[r] + bnB), orow + colB + l16);
        }
    }
}

extern "C" void kernel_launch(void* const* d_in, const int* in_sizes, int n_in,
                              void* d_out, int out_size, void* d_ws, size_t ws_size,
                              hipStream_t stream) {
    (void)in_sizes; (void)n_in; (void)d_ws; (void)ws_size; (void)out_size;
    const float*         outp = (const float*)d_in[0];         // output [B,T,H]
    const float*         ctx  = (const float*)d_in[1];         // context [B,S,H]
    const unsigned char* mask = (const unsigned char*)d_in[2]; // mask [B,T,S] bool
    const float*         W    = (const float*)d_in[3];         // W_out [H,2H]
    const float*         bias = (const float*)d_in[4];         // b_out [H]

    float* out      = (float*)d_out;                 // out [B,T,H] first
    float* attn_out = out + (size_t)BB * TT * HH;    // then attn [B,T,S]

    dim3 grid(TT / 16, BB);
    dim3 block(256);
    AttentionMechanism_45406394254132_kernel<<<grid, block, 0, stream>>>(
        outp, ctx, mask, W, bias, out, attn_out);
}